// BEVEncoderLayer_71373766525401
// MI455X (gfx1250) — compile-verified
//
#include <hip/hip_runtime.h>

#define EMBED 256
#define HEADS 8
#define DH 32
#define POINTS 4
#define LEVELS 2
#define NCAMS 6
#define BEVW 100
#define NQ (BEVW * BEVW)
#define KC 128              // K-chunk staged in LDS per iteration

typedef float v2f __attribute__((ext_vector_type(2)));
typedef float v4f __attribute__((ext_vector_type(4), aligned(16)));
typedef float v8f __attribute__((ext_vector_type(8)));
typedef int   v4i_async __attribute__((vector_size(16)));   // matches builtin proto
typedef __attribute__((address_space(1))) v4i_async* gptr4_t;
typedef __attribute__((address_space(3))) v4i_async* lptr4_t;

#if defined(__gfx1250__) && __has_builtin(__builtin_amdgcn_global_load_async_to_lds_b128)
#define USE_ASYNC_LDS 1
#endif

// ---------------------------------------------------------------------------
// fp32 WMMA GEMM: C[M,Nn] = A[M,K] @ W[K,Nn] + bias, optional ReLU.
// Block = 8 waves; each block owns one 16-wide N strip and 8 M-tiles.
// B chunk (KC x 16) is staged in LDS once per block via async-to-LDS and
// reused by all 8 waves; WMMA = V_WMMA_F32_16X16X4_F32.
// FEAT=true: A is in NCHW camera-feature layout (Cimg, K, HW):
//   row m = c*HW + yx  ->  A(m,k) = A[(c*K + k)*HW + yx].
// ---------------------------------------------------------------------------
template <bool FEAT>
__global__ void gemm_wmma_f32(const float* __restrict__ A,
                              const float* __restrict__ W,
                              const float* __restrict__ bias,
                              float* __restrict__ C,
                              int M, int K, int Nn, int HW, int relu)
{
    __shared__ __attribute__((aligned(16))) float ldsB[KC * 16];

    const int lane   = threadIdx.x & 31;
    const int wv     = threadIdx.x >> 5;          // 0..7
    const int tilesN = Nn >> 4;
    const int strip  = blockIdx.x / tilesN;
    const int tn     = (blockIdx.x % tilesN) << 4;
    const int tm     = (strip * 8 + wv) << 4;
    const bool active = (tm < M);                 // wave-uniform

    const int half = lane >> 4;                   // 0 -> K {0,1}, 1 -> K {2,3}
    const int l15  = lane & 15;

    size_t aBase = 0;
    if (active) {
        const int m = tm + l15;
        if (FEAT) {
            const int c  = m / HW;
            const int yx = m - c * HW;
            aBase = (size_t)c * K * HW + yx;
        } else {
            aBase = (size_t)m * K;
        }
    }

    v8f acc = {0.f, 0.f, 0.f, 0.f, 0.f, 0.f, 0.f, 0.f};

    for (int k0 = 0; k0 < K; k0 += KC) {
        __syncthreads();                           // ldsB free to overwrite
        // ---- cooperative stage of B chunk: KC*16 floats = 512 float4s ----
        for (int e = threadIdx.x; e < (KC * 16) / 4; e += blockDim.x) {
            const int kloc = e >> 2;
            const int cg   = e & 3;
            const float* src = W + (size_t)(k0 + kloc) * Nn + tn + cg * 4;
            float* dst = &ldsB[kloc * 16 + cg * 4];
#if USE_ASYNC_LDS
            __builtin_amdgcn_global_load_async_to_lds_b128(
                (gptr4_t)(float*)src, (lptr4_t)dst, 0, 0);
#else
            *(v4f*)dst = *(const v4f*)src;
#endif
        }
#if USE_ASYNC_LDS
#if __has_builtin(__builtin_amdgcn_s_wait_asynccnt)
        __builtin_amdgcn_s_wait_asynccnt(0);
#else
        asm volatile("s_wait_asynccnt 0x0" ::: "memory");
#endif
#endif
        __syncthreads();                           // chunk visible to all waves

        if (active) {
            const float* __restrict__ aPtr = A + aBase + (FEAT ? 0 : (size_t)k0);
            #pragma unroll 8
            for (int k = 0; k < KC; k += 4) {
                const int kk = k + 2 * half;
                v2f a;
                if (FEAT) {
                    a.x = aPtr[(size_t)(k0 + kk) * HW];
                    a.y = aPtr[(size_t)(k0 + kk + 1) * HW];
                } else {
                    a = *(const v2f*)(aPtr + kk);           // single b64 load
                }
                v2f b;
                b.x = ldsB[kk * 16 + l15];
                b.y = ldsB[(kk + 1) * 16 + l15];            // ds_load_2addr
                acc = __builtin_amdgcn_wmma_f32_16x16x4_f32(false, a, false, b,
                                                            (short)0, acc, false, false);
            }
        }
    }

    if (active) {
        const float bs    = bias ? bias[tn + l15] : 0.f;
        const int   mbase = tm + half * 8;
        #pragma unroll
        for (int r = 0; r < 8; ++r) {
            float v = acc[r] + bs;
            if (relu) v = fmaxf(v, 0.f);
            C[(size_t)(mbase + r) * Nn + tn + l15] = v;
        }
    }
}

// ---------------------------------------------------------------------------
// Softmax over contiguous groups of length L (L = 4 temporal, 8 spatial).
// ---------------------------------------------------------------------------
__global__ void softmax_groups(float* __restrict__ a, int ngroups, int L)
{
    int g = blockIdx.x * blockDim.x + threadIdx.x;
    if (g >= ngroups) return;
    float* p = a + (size_t)g * L;
    float mx = p[0];
    for (int i = 1; i < L; ++i) mx = fmaxf(mx, p[i]);
    float s = 0.f;
    for (int i = 0; i < L; ++i) { float e = __expf(p[i] - mx); p[i] = e; s += e; }
    float inv = 1.f / s;
    for (int i = 0; i < L; ++i) p[i] *= inv;
}

// ---------------------------------------------------------------------------
// Temporal deformable sampling: one wave per (query n, head h); lane = channel.
// ---------------------------------------------------------------------------
__device__ __forceinline__ float tap_bev(const float* __restrict__ v,
                                         int yi, int xi, int chan)
{
    if (xi < 0 || xi >= BEVW || yi < 0 || yi >= BEVW) return 0.f;
    return v[(size_t)(yi * BEVW + xi) * EMBED + chan];
}

__global__ void temporal_sample(const float* __restrict__ offt,
                                const float* __restrict__ attnt,
                                const float* __restrict__ vt,
                                float* __restrict__ tsa)
{
    const int lane = threadIdx.x & 31;
    const int w    = (blockIdx.x * blockDim.x + threadIdx.x) >> 5;
    if (w >= NQ * HEADS) return;
    const int n    = w >> 3;
    const int h    = w & 7;
    const int chan = h * DH + lane;
    const float refx = ((n % BEVW) + 0.5f) * (1.f / BEVW);
    const float refy = ((n / BEVW) + 0.5f) * (1.f / BEVW);
    float acc = 0.f;
    #pragma unroll
    for (int p = 0; p < POINTS; ++p) {
        const float ox = offt[(size_t)n * 64 + h * 8 + p * 2 + 0];
        const float oy = offt[(size_t)n * 64 + h * 8 + p * 2 + 1];
        const float x  = (refx + ox * (1.f / BEVW)) * BEVW - 0.5f;
        const float y  = (refy + oy * (1.f / BEVW)) * BEVW - 0.5f;
        const float x0 = floorf(x), y0 = floorf(y);
        const float wx = x - x0,    wy = y - y0;
        const int   xi = (int)x0,   yi = (int)y0;
        const float s =
              tap_bev(vt, yi,     xi,     chan) * (1.f - wx) * (1.f - wy)
            + tap_bev(vt, yi,     xi + 1, chan) * wx         * (1.f - wy)
            + tap_bev(vt, yi + 1, xi,     chan) * (1.f - wx) * wy
            + tap_bev(vt, yi + 1, xi + 1, chan) * wx         * wy;
        acc += attnt[(size_t)n * 32 + h * 4 + p] * s;
    }
    tsa[(size_t)n * EMBED + chan] = acc;
}

// ---------------------------------------------------------------------------
// Camera projection: thread per (cam c, query n).
// ---------------------------------------------------------------------------
__global__ void cam_project(const float* __restrict__ rp,
                            const float* __restrict__ Km,
                            const float* __restrict__ Em,
                            float* __restrict__ refc,
                            float* __restrict__ maskz)
{
    int t = blockIdx.x * blockDim.x + threadIdx.x;
    if (t >= NCAMS * NQ) return;
    int c = t / NQ;
    int n = t - c * NQ;
    float px = rp[n * 3 + 0], py = rp[n * 3 + 1], pz = rp[n * 3 + 2];
    const float* E = Em + c * 16;
    float c0 = E[0] * px + E[1] * py + E[2]  * pz + E[3];
    float c1 = E[4] * px + E[5] * py + E[6]  * pz + E[7];
    float c2 = E[8] * px + E[9] * py + E[10] * pz + E[11];
    const float* Kc = Km + c * 9;
    float p0 = Kc[0] * c0 + Kc[1] * c1 + Kc[2] * c2;
    float p1 = Kc[3] * c0 + Kc[4] * c1 + Kc[5] * c2;
    float p2 = Kc[6] * c0 + Kc[7] * c1 + Kc[8] * c2;
    float z  = p2;
    float zd = fmaxf(z, 1e-5f);
    float rx = p0 / zd * (1.f / 704.f);
    float ry = p1 / zd * (1.f / 256.f);
    int m = (z > 1e-5f) && (rx >= 0.f) && (rx <= 1.f) && (ry >= 0.f) && (ry <= 1.f);
    refc[(size_t)t * 2 + 0] = rx;
    refc[(size_t)t * 2 + 1] = ry;
    maskz[t] = m ? 1.f : 0.f;
}

// ---------------------------------------------------------------------------
// Spatial deformable sampling: one wave per (n, h); lane = channel.
// ---------------------------------------------------------------------------
__global__ void spatial_sample(const float* __restrict__ offs,
                               const float* __restrict__ attns,
                               const float* __restrict__ refc,
                               const float* __restrict__ maskz,
                               const float* __restrict__ vl0,
                               const float* __restrict__ vl1,
                               float* __restrict__ sca)
{
    const int lane = threadIdx.x & 31;
    const int w    = (blockIdx.x * blockDim.x + threadIdx.x) >> 5;
    if (w >= NQ * HEADS) return;
    const int n = w >> 3;
    const int h = w & 7;
    const int chan = h * DH + lane;

    float cnt = 0.f;
    #pragma unroll
    for (int c = 0; c < NCAMS; ++c) cnt += maskz[(size_t)c * NQ + n];
    cnt = fmaxf(cnt, 1.f);

    float acc = 0.f;
    for (int c = 0; c < NCAMS; ++c) {
        if (maskz[(size_t)c * NQ + n] < 0.5f) continue;   // wave-uniform
        const float rx = refc[((size_t)c * NQ + n) * 2 + 0];
        const float ry = refc[((size_t)c * NQ + n) * 2 + 1];
        #pragma unroll
        for (int l = 0; l < LEVELS; ++l) {
            const int Hl = l ? 16 : 32;
            const int Wl = l ? 44 : 88;
            const float* vl = l ? vl1 : vl0;
            const size_t planeBase = (size_t)c * Hl * Wl;
            #pragma unroll
            for (int p = 0; p < POINTS; ++p) {
                const float ox = offs[(size_t)n * 128 + h * 16 + l * 8 + p * 2 + 0];
                const float oy = offs[(size_t)n * 128 + h * 16 + l * 8 + p * 2 + 1];
                const float x  = (rx + ox / Wl) * Wl - 0.5f;
                const float y  = (ry + oy / Hl) * Hl - 0.5f;
                const float x0 = floorf(x), y0 = floorf(y);
                const float wx = x - x0,    wy = y - y0;
                const int   xi = (int)x0,   yi = (int)y0;
                float s = 0.f;
                #pragma unroll
                for (int ty = 0; ty < 2; ++ty) {
                    #pragma unroll
                    for (int tx = 0; tx < 2; ++tx) {
                        const int yy = yi + ty, xx = xi + tx;
                        const float wgt = (tx ? wx : 1.f - wx) * (ty ? wy : 1.f - wy);
                        if (xx >= 0 && xx < Wl && yy >= 0 && yy < Hl)
                            s += vl[(planeBase + (size_t)yy * Wl + xx) * EMBED + chan] * wgt;
                    }
                }
                acc += attns[(size_t)n * 64 + h * 8 + l * 4 + p] * s;
            }
        }
    }
    sca[(size_t)n * EMBED + h * DH + lane] = acc / cnt;
}

// ---------------------------------------------------------------------------
// LayerNorm(res + x) * s + b : one wave per row of 256.
// ---------------------------------------------------------------------------
__global__ void layernorm_residual(const float* __restrict__ res,
                                   const float* __restrict__ x,
                                   const float* __restrict__ s,
                                   const float* __restrict__ b,
                                   float* __restrict__ out, int rows)
{
    const int lane = threadIdx.x & 31;
    const int row  = (blockIdx.x * blockDim.x + threadIdx.x) >> 5;
    if (row >= rows) return;
    float vals[8];
    float sum = 0.f;
    #pragma unroll
    for (int r = 0; r < 8; ++r) {
        const int d = lane + r * 32;
        const float v = res[(size_t)row * EMBED + d] + x[(size_t)row * EMBED + d];
        vals[r] = v; sum += v;
    }
    #pragma unroll
    for (int m = 16; m >= 1; m >>= 1) sum += __shfl_xor(sum, m, 32);
    const float mean = sum * (1.f / EMBED);
    float var = 0.f;
    #pragma unroll
    for (int r = 0; r < 8; ++r) { const float d = vals[r] - mean; var += d * d; }
    #pragma unroll
    for (int m = 16; m >= 1; m >>= 1) var += __shfl_xor(var, m, 32);
    var *= (1.f / EMBED);
    const float inv = rsqrtf(var + 1e-5f);
    #pragma unroll
    for (int r = 0; r < 8; ++r) {
        const int d = lane + r * 32;
        out[(size_t)row * EMBED + d] = (vals[r] - mean) * inv * s[d] + b[d];
    }
}

// ---------------------------------------------------------------------------
extern "C" void kernel_launch(void* const* d_in, const int* in_sizes, int n_in,
                              void* d_out, int out_size, void* d_ws, size_t ws_size,
                              hipStream_t stream)
{
    const float* bq     = (const float*)d_in[0];
    const float* pb     = (const float*)d_in[1];
    const float* feat0  = (const float*)d_in[2];   // (6,256,32,88)
    const float* feat1  = (const float*)d_in[3];   // (6,256,16,44)
    const float* rp     = (const float*)d_in[4];
    const float* Kc     = (const float*)d_in[5];
    const float* Ec     = (const float*)d_in[6];
    // d_in[7] = spatial_shapes (unused; dims are compile-time constants)
    const float* Wv_t    = (const float*)d_in[8];
    const float* bv_t    = (const float*)d_in[9];
    const float* Woff_t  = (const float*)d_in[10];
    const float* boff_t  = (const float*)d_in[11];
    const float* Wattn_t = (const float*)d_in[12];
    const float* battn_t = (const float*)d_in[13];
    const float* Wout_t  = (const float*)d_in[14];
    const float* bout_t  = (const float*)d_in[15];
    const float* ln1_s   = (const float*)d_in[16];
    const float* ln1_b   = (const float*)d_in[17];
    const float* Wv_s    = (const float*)d_in[18];
    const float* bv_s    = (const float*)d_in[19];
    const float* Woff_s  = (const float*)d_in[20];
    const float* boff_s  = (const float*)d_in[21];
    const float* Wattn_s = (const float*)d_in[22];
    const float* battn_s = (const float*)d_in[23];
    const float* Wout_s  = (const float*)d_in[24];
    const float* bout_s  = (const float*)d_in[25];
    const float* ln2_s   = (const float*)d_in[26];
    const float* ln2_b   = (const float*)d_in[27];
    const float* W1      = (const float*)d_in[28];
    const float* b1      = (const float*)d_in[29];
    const float* W2      = (const float*)d_in[30];
    const float* b2      = (const float*)d_in[31];
    const float* ln3_s   = (const float*)d_in[32];
    const float* ln3_b   = (const float*)d_in[33];

    const int N = NQ, D = EMBED;
    const int M0 = NCAMS * 32 * 88;   // 16896
    const int M1 = NCAMS * 16 * 44;   // 4224

    // ---- workspace layout (floats), with aliasing for dead buffers ----
    float* ws = (float*)d_ws;
    size_t o = 0;
    float* q1    = ws + o; o += (size_t)N * D;
    float* q2    = ws + o; o += (size_t)N * D;
    float* sca   = ws + o; o += (size_t)N * D;
    float* tsa   = ws + o; o += (size_t)N * D;       // later reused as FFN output f
    float* tmp   = ws + o; o += (size_t)N * D;
    float* offt  = ws + o; o += (size_t)N * 64;
    float* attnt = ws + o; o += (size_t)N * 32;
    float* refc  = ws + o; o += (size_t)NCAMS * N * 2;
    float* maskz = ws + o; o += (size_t)NCAMS * N;
    float* offs_ = ws + o; o += (size_t)N * 128;     // FFN hidden starts here later
    float* attns_= ws + o; o += (size_t)N * 64;
    float* vt    = ws + o; o += (size_t)N * D;       // later reused as vl1
    float* vl0   = ws + o; o += (size_t)M0 * D;
    size_t hidNeed = (size_t)N * 1024;
    size_t have = (size_t)((ws + o) - offs_);
    if (have < hidNeed) o += hidNeed - have;
    float* hid = offs_;   // aliases offs_/attns_/vt/vl0 (dead by FFN time)
    float* f   = tsa;     // tsa dead after its GEMM
    float* vl1 = vt;      // vt dead after temporal sampling

    auto gemmBlocks = [](int M, int Nn) {
        int strips = ((M >> 4) + 7) >> 3;            // 8 M-tiles per block
        return strips * (Nn >> 4);
    };
    const int TB = 256;

    // ---- 1. temporal value projection: vt = pb @ Wv_t + bv_t
    gemm_wmma_f32<false><<<gemmBlocks(N, D), TB, 0, stream>>>(pb, Wv_t, bv_t, vt, N, D, D, 0, 0);
    // ---- 2-3. temporal offsets / attention logits
    gemm_wmma_f32<false><<<gemmBlocks(N, 64), TB, 0, stream>>>(bq, Woff_t, boff_t, offt, N, D, 64, 0, 0);
    gemm_wmma_f32<false><<<gemmBlocks(N, 32), TB, 0, stream>>>(bq, Wattn_t, battn_t, attnt, N, D, 32, 0, 0);
    softmax_groups<<<(N * HEADS + TB - 1) / TB, TB, 0, stream>>>(attnt, N * HEADS, POINTS);
    // ---- 4. temporal deformable sampling -> tsa
    temporal_sample<<<(N * HEADS) / 8, TB, 0, stream>>>(offt, attnt, vt, tsa);
    // ---- 5-6. output proj + residual LN -> q1
    gemm_wmma_f32<false><<<gemmBlocks(N, D), TB, 0, stream>>>(tsa, Wout_t, bout_t, tmp, N, D, D, 0, 0);
    layernorm_residual<<<(N + 7) / 8, TB, 0, stream>>>(bq, tmp, ln1_s, ln1_b, q1, N);
    // ---- 7. camera projection
    cam_project<<<(NCAMS * N + TB - 1) / TB, TB, 0, stream>>>(rp, Kc, Ec, refc, maskz);
    // ---- 8. spatial offsets / attention
    gemm_wmma_f32<false><<<gemmBlocks(N, 128), TB, 0, stream>>>(q1, Woff_s, boff_s, offs_, N, D, 128, 0, 0);
    gemm_wmma_f32<false><<<gemmBlocks(N, 64), TB, 0, stream>>>(q1, Wattn_s, battn_s, attns_, N, D, 64, 0, 0);
    softmax_groups<<<(N * HEADS + TB - 1) / TB, TB, 0, stream>>>(attns_, N * HEADS, LEVELS * POINTS);
    // ---- 9. per-level value projections, gathering A straight from NCHW feats
    gemm_wmma_f32<true><<<gemmBlocks(M0, D), TB, 0, stream>>>(feat0, Wv_s, bv_s, vl0, M0, D, D, 32 * 88, 0);
    gemm_wmma_f32<true><<<gemmBlocks(M1, D), TB, 0, stream>>>(feat1, Wv_s, bv_s, vl1, M1, D, D, 16 * 44, 0);
    // ---- 10. spatial deformable sampling -> sca
    spatial_sample<<<(N * HEADS) / 8, TB, 0, stream>>>(offs_, attns_, refc, maskz, vl0, vl1, sca);
    // ---- 11. output proj + residual LN -> q2
    gemm_wmma_f32<false><<<gemmBlocks(N, D), TB, 0, stream>>>(sca, Wout_s, bout_s, tmp, N, D, D, 0, 0);
    layernorm_residual<<<(N + 7) / 8, TB, 0, stream>>>(q1, tmp, ln2_s, ln2_b, q2, N);
    // ---- 12. FFN with fused ReLU, then residual LN -> d_out
    gemm_wmma_f32<false><<<gemmBlocks(N, 1024), TB, 0, stream>>>(q2, W1, b1, hid, N, D, 1024, 0, 1);
    gemm_wmma_f32<false><<<gemmBlocks(N, D), TB, 0, stream>>>(hid, W2, b2, f, N, 1024, D, 0, 0);
    layernorm_residual<<<(N + 7) / 8, TB, 0, stream>>>(q2, f, ln3_s, ln3_b, (float*)d_out, N);
}